// ValueDecoder_16140487098928
// MI455X (gfx1250) — compile-verified
//
#include <hip/hip_runtime.h>

typedef float v2f __attribute__((ext_vector_type(2)));
typedef float v8f __attribute__((ext_vector_type(8)));

#define GS_B 32
#define GS_N 1024
#define GS_D 256
#define GS_ITERS 20

// ---------------------------------------------------------------------------
// Wave32 butterfly reductions (no barriers, no LDS traffic).
// ---------------------------------------------------------------------------
__device__ __forceinline__ float wave_max_f32(float x) {
#pragma unroll
  for (int off = 16; off >= 1; off >>= 1) x = fmaxf(x, __shfl_xor(x, off, 32));
  return x;
}
__device__ __forceinline__ float wave_sum_f32(float x) {
#pragma unroll
  for (int off = 16; off >= 1; off >>= 1) x += __shfl_xor(x, off, 32);
  return x;
}

// ---------------------------------------------------------------------------
// Kernel 1: scalar_scores = scores @ W + b via V_WMMA_F32_16X16X4_F32.
// One wave computes a 16-row tile; W packed into B-matrix column N=0 with a
// branchless per-lane mask (keeps EXEC all-1s through the WMMA loop).
// A layout (f32 16x4): lanes 0-15 hold (K=0,K=1), lanes 16-31 hold (K=2,K=3).
// D layout: VGPR r -> M=r (lanes 0-15, N=lane) / M=r+8 (lanes 16-31).
// ---------------------------------------------------------------------------
__global__ __launch_bounds__(32)
void gemv_wmma_kernel(const float* __restrict__ scores,
                      const float* __restrict__ W,
                      const float* __restrict__ bias,
                      float* __restrict__ S) {
  const int row_base = blockIdx.x * 16;      // into flattened (B*N)
  const int lane = threadIdx.x;              // 0..31
  const int half = lane >> 4;                // 0 or 1
  const int m    = lane & 15;
  const float msel = (m == 0) ? 1.0f : 0.0f; // N==0 column carries W

  const float* arow = scores + (size_t)(row_base + m) * GS_D + 2 * half;
  const float* wrow = W + 2 * half;

  v8f acc = {0.f, 0.f, 0.f, 0.f, 0.f, 0.f, 0.f, 0.f};

  for (int k = 0; k < GS_D; k += 4) {
    v2f a;
    a.x = arow[k];
    a.y = arow[k + 1];
    v2f bmat;
    bmat.x = msel * wrow[k];
    bmat.y = msel * wrow[k + 1];
    acc = __builtin_amdgcn_wmma_f32_16x16x4_f32(
        /*neg_a=*/false, a, /*neg_b=*/false, bmat,
        /*c_mod=*/(short)0, acc, /*reuse_a=*/false, /*reuse_b=*/false);
  }

  const float bb = bias[0];
  if (m == 0) {  // N == 0 column holds the dot products
#pragma unroll
    for (int r = 0; r < 8; ++r) {
      S[row_base + r + 8 * half] = acc[r] + bb;
    }
  }
}

// ---------------------------------------------------------------------------
// Zero-init R and C (contiguous in workspace).
// ---------------------------------------------------------------------------
__global__ void zero_kernel(float* __restrict__ p, int n) {
  int i = blockIdx.x * blockDim.x + threadIdx.x;
  if (i < n) p[i] = 0.0f;
}

// ---------------------------------------------------------------------------
// Row pass: R[b,i] = LSE_j( S[b,i] + g[b,i,j] - C[b,j] ).
// One block per row; 256 threads x 4 elements. Wave shuffle reduce + 8-way
// LDS combine (2 barriers total instead of a 16-barrier tree).
// ---------------------------------------------------------------------------
__global__ __launch_bounds__(256)
void row_lse_kernel(const float* __restrict__ g, const float* __restrict__ S,
                    const float* __restrict__ C, float* __restrict__ R) {
  const int row = blockIdx.x;       // b*N + i
  const int b   = row >> 10;        // / GS_N
  const int t   = threadIdx.x;
  const int wid = t >> 5;           // wave id 0..7
  const float s = S[row];
  const float* gr = g + (size_t)row * GS_N;
  const float* Cb = C + b * GS_N;

  float v[4];
  float m = -3.402823466e+38f;
#pragma unroll
  for (int q = 0; q < 4; ++q) {
    const int j = t + q * 256;
    v[q] = s + gr[j] - Cb[j];
    m = fmaxf(m, v[q]);
  }

  __shared__ float wredm[8];
  __shared__ float wreds[8];

  const float wm = wave_max_f32(m);
  if ((t & 31) == 0) wredm[wid] = wm;
  __syncthreads();

  float bm = wredm[0];
#pragma unroll
  for (int w = 1; w < 8; ++w) bm = fmaxf(bm, wredm[w]);

  float lsum = 0.0f;
#pragma unroll
  for (int q = 0; q < 4; ++q) lsum += __expf(v[q] - bm);
  const float wsum = wave_sum_f32(lsum);
  if ((t & 31) == 0) wreds[wid] = wsum;
  __syncthreads();

  if (t == 0) {
    float tot = wreds[0];
#pragma unroll
    for (int w = 1; w < 8; ++w) tot += wreds[w];
    R[row] = bm + __logf(tot);
  }
}

// ---------------------------------------------------------------------------
// Column pass: C[b,j] = LSE_i( S[b,i] + g[b,i,j] - R[b,i] ).
// Block = 64 columns x 4 row-groups (256 rows each). Online LSE per thread,
// 4-way LDS merge. (S - R) staged in LDS; global_prefetch_b8 for upcoming rows.
// ---------------------------------------------------------------------------
__global__ __launch_bounds__(256)
void col_lse_kernel(const float* __restrict__ g, const float* __restrict__ S,
                    const float* __restrict__ R, float* __restrict__ C) {
  const int blk = blockIdx.x;
  const int b   = blk >> 4;          // 16 column-groups of 64 per batch
  const int jg  = blk & 15;
  const int t   = threadIdx.x;
  const int jj  = t & 63;
  const int rr  = t >> 6;            // 0..3 row-group
  const int j   = jg * 64 + jj;

  const float* gb = g + (size_t)b * GS_N * GS_N;
  const float* Sb = S + b * GS_N;
  const float* Rb = R + b * GS_N;

  __shared__ float shSR[GS_N];
  for (int i = t; i < GS_N; i += 256) shSR[i] = Sb[i] - Rb[i];
  __syncthreads();

  float m = -3.402823466e+38f;
  float ssum = 0.0f;
  const int i0 = rr * 256;
  for (int i = i0; i < i0 + 256; ++i) {
    __builtin_prefetch(gb + (size_t)(i + 8) * GS_N + j, 0, 1);
    const float v = shSR[i] + gb[(size_t)i * GS_N + j];
    const float mn = fmaxf(m, v);
    ssum = ssum * __expf(m - mn) + __expf(v - mn);
    m = mn;
  }

  __shared__ float sm[256];
  __shared__ float ss[256];
  sm[t] = m;
  ss[t] = ssum;
  __syncthreads();
  if (rr == 0) {
    float M = sm[jj], Q = ss[jj];
#pragma unroll
    for (int k = 1; k < 4; ++k) {
      const float m2 = sm[jj + 64 * k];
      const float s2 = ss[jj + 64 * k];
      const float Mn = fmaxf(M, m2);
      Q = Q * __expf(M - Mn) + s2 * __expf(m2 - Mn);
      M = Mn;
    }
    C[b * GS_N + j] = M + __logf(Q);
  }
}

// ---------------------------------------------------------------------------
// Final: out[b,i,j] = exp( S[b,i] + g[b,i,j] - R[b,i] - C[b,j] ).
// One block per row, float4 vectorized (b128 loads/stores).
// ---------------------------------------------------------------------------
__global__ __launch_bounds__(256)
void final_exp_kernel(const float* __restrict__ g, const float* __restrict__ S,
                      const float* __restrict__ R, const float* __restrict__ C,
                      float* __restrict__ out) {
  const int row = blockIdx.x;       // b*N + i
  const int b   = row >> 10;
  const int t   = threadIdx.x;
  const float sr = S[row] - R[row];
  const float4* gv = (const float4*)(g + (size_t)row * GS_N);
  const float4* cv = (const float4*)(C + b * GS_N);
  float4* ov = (float4*)(out + (size_t)row * GS_N);

  const float4 gq = gv[t];
  const float4 cq = cv[t];
  float4 o;
  o.x = __expf(sr + gq.x - cq.x);
  o.y = __expf(sr + gq.y - cq.y);
  o.z = __expf(sr + gq.z - cq.z);
  o.w = __expf(sr + gq.w - cq.w);
  ov[t] = o;
}

// ---------------------------------------------------------------------------
extern "C" void kernel_launch(void* const* d_in, const int* in_sizes, int n_in,
                              void* d_out, int out_size, void* d_ws, size_t ws_size,
                              hipStream_t stream) {
  (void)in_sizes; (void)n_in; (void)out_size; (void)ws_size;

  const float* scores = (const float*)d_in[0];   // (B, N, D)
  const float* W      = (const float*)d_in[1];   // (D, 1)
  const float* bias   = (const float*)d_in[2];   // (1,)
  const float* g      = (const float*)d_in[3];   // (B, N, N) gumbel noise
  float* out = (float*)d_out;                    // (B, N, N)

  float* S = (float*)d_ws;                       // (B*N) scalar scores
  float* R = S + GS_B * GS_N;                    // (B*N) row offsets
  float* C = R + GS_B * GS_N;                    // (B*N) col offsets

  // scalar_scores via WMMA
  gemv_wmma_kernel<<<GS_B * GS_N / 16, 32, 0, stream>>>(scores, W, bias, S);
  // R = C = 0 (contiguous)
  zero_kernel<<<(2 * GS_B * GS_N) / 256, 256, 0, stream>>>(R, 2 * GS_B * GS_N);

  for (int it = 0; it < GS_ITERS; ++it) {
    row_lse_kernel<<<GS_B * GS_N, 256, 0, stream>>>(g, S, C, R);
    col_lse_kernel<<<GS_B * (GS_N / 64), 256, 0, stream>>>(g, S, R, C);
  }

  final_exp_kernel<<<GS_B * GS_N, 256, 0, stream>>>(g, S, R, C, out);
}